// MinimalLBS_6880537608623
// MI455X (gfx1250) — compile-verified
//
#include <hip/hip_runtime.h>

typedef __attribute__((ext_vector_type(2))) float v2f;
typedef __attribute__((ext_vector_type(8))) float v8f;

#define B_  32
#define T_  128
#define JB_ 21
#define J_  52
#define N_  2048
#define L_  16
#define M_  (N_*3)      // 6144 flattened (n,xyz) columns
#define PFD 189         // Jb*9
#define PFP 192         // padded pose-feature length (48 K-steps of 4)

#define WMMA_F32(a, bfrag, c) \
    __builtin_amdgcn_wmma_f32_16x16x4_f32(false, (a), false, (bfrag), (short)0, (c), false, false)

// ---------------------------------------------------------------------------
// Kernel 1: axis-angle -> (R - I) pose features, padded to 192 with zeros.
// ---------------------------------------------------------------------------
__global__ void pf_kernel(const float* __restrict__ pose_body,
                          float* __restrict__ pf) {
    int idx = blockIdx.x * blockDim.x + threadIdx.x;
    if (idx >= B_ * T_ * 22) return;
    int j  = idx % 22;
    int bt = idx / 22;
    float* pfrow = pf + (size_t)bt * PFP;
    if (j == 21) { pfrow[189] = 0.f; pfrow[190] = 0.f; pfrow[191] = 0.f; return; }
    const float* aa = pose_body + (size_t)bt * (JB_*3) + j*3;
    float x = aa[0], y = aa[1], z = aa[2];
    float ang = sqrtf(x*x + y*y + z*z);
    float inv = 1.0f / fmaxf(ang, 1e-8f);
    float ux = x*inv, uy = y*inv, uz = z*inv;
    float s = sinf(ang), c = cosf(ang), oc = 1.0f - c;
    float* o = pfrow + j*9;
    o[0] = c + oc*ux*ux - 1.0f;
    o[1] = -s*uz + oc*ux*uy;
    o[2] =  s*uy + oc*ux*uz;
    o[3] =  s*uz + oc*uy*ux;
    o[4] = c + oc*uy*uy - 1.0f;
    o[5] = -s*ux + oc*uy*uz;
    o[6] = -s*uy + oc*uz*ux;
    o[7] =  s*ux + oc*uz*uy;
    o[8] = c + oc*uz*uz - 1.0f;
}

// ---------------------------------------------------------------------------
// Kernel 2: v_posed = pf @ posedirs + betas @ shapedirs_v + v_template.
// Each wave register-blocks 2 t-tiles x 2 m-tiles (4 WMMA accumulators):
// every A-fragment and every B-fragment feeds 2 WMMAs -> half the L2 traffic.
// grid = (6144/32/8, T/32, B) = (24, 4, 32), 256 threads (8 waves).
// ---------------------------------------------------------------------------
__global__ void vposed_kernel(const float* __restrict__ pf,        // (B,T,192)
                              const float* __restrict__ betas,     // (B,T,16)
                              const float* __restrict__ posedirs,  // (B,189,6144)
                              const float* __restrict__ shapedirs, // (B,6144,16)
                              const float* __restrict__ v_template,// (B,6144)
                              float* __restrict__ v_posed) {       // (B,T,6144)
    int lane  = threadIdx.x & 31;
    int w     = threadIdx.x >> 5;
    int b     = blockIdx.z;
    int t0    = blockIdx.y * 32;                 // t-tiles at t0, t0+16
    int m0    = (blockIdx.x * 8 + w) * 32;       // m-tiles at m0, m0+16
    int r15   = lane & 15;                       // A row (t) / B col (m)
    int khalf = (lane >> 4) * 2;                 // lanes 16..31 hold K+2,K+3

    v8f acc00 = {0.f,0.f,0.f,0.f,0.f,0.f,0.f,0.f};
    v8f acc01 = acc00, acc10 = acc00, acc11 = acc00;

    // --- posedirs GEMM: K = 189 padded to 192 (pf zero-padded) ---
    const float* pfA0 = pf + (size_t)(b*T_ + t0 + r15) * PFP + khalf;
    const float* pfA1 = pfA0 + 16 * PFP;
    const float* pdB0 = posedirs + (size_t)b * (PFD * (size_t)M_) + (m0 + r15);
    const float* pdB1 = pdB0 + 16;
    #pragma unroll
    for (int kk = 0; kk < 48; ++kk) {
        int k0 = kk * 4;
        v2f a0 = *(const v2f*)(pfA0 + k0);
        v2f a1 = *(const v2f*)(pfA1 + k0);
        int r0  = k0 + khalf;
        int r0c = (r0     > 188) ? 188 : r0;     // clamp; pf==0 past 188
        int r1c = (r0 + 1 > 188) ? 188 : r0 + 1;
        v2f b0, b1;
        b0.x = pdB0[(size_t)r0c * M_];  b0.y = pdB0[(size_t)r1c * M_];
        b1.x = pdB1[(size_t)r0c * M_];  b1.y = pdB1[(size_t)r1c * M_];
        acc00 = WMMA_F32(a0, b0, acc00);
        acc01 = WMMA_F32(a0, b1, acc01);
        acc10 = WMMA_F32(a1, b0, acc10);
        acc11 = WMMA_F32(a1, b1, acc11);
    }

    // --- shapedirs GEMM: K = 16 ---
    const float* beA0 = betas     + (size_t)(b*T_ + t0 + r15) * L_ + khalf;
    const float* beA1 = beA0 + 16 * L_;
    const float* sdB0 = shapedirs + (size_t)(b*M_ + m0 + r15) * L_ + khalf;
    const float* sdB1 = sdB0 + 16 * L_;
    #pragma unroll
    for (int kk = 0; kk < 4; ++kk) {
        v2f a0 = *(const v2f*)(beA0 + kk*4);
        v2f a1 = *(const v2f*)(beA1 + kk*4);
        v2f b0 = *(const v2f*)(sdB0 + kk*4);
        v2f b1 = *(const v2f*)(sdB1 + kk*4);
        acc00 = WMMA_F32(a0, b0, acc00);
        acc01 = WMMA_F32(a0, b1, acc01);
        acc10 = WMMA_F32(a1, b0, acc10);
        acc11 = WMMA_F32(a1, b1, acc11);
    }

    // --- + v_template, store ---
    float vt0 = v_template[(size_t)b * M_ + m0 + r15];
    float vt1 = v_template[(size_t)b * M_ + m0 + 16 + r15];
    #pragma unroll
    for (int r = 0; r < 8; ++r) {
        int ta = t0 + r + (lane >> 4) * 8;
        int tb = ta + 16;
        float* oa = v_posed + (size_t)(b*T_ + ta) * M_ + m0 + r15;
        float* ob = v_posed + (size_t)(b*T_ + tb) * M_ + m0 + r15;
        oa[0]  = acc00[r] + vt0;
        oa[16] = acc01[r] + vt1;
        ob[0]  = acc10[r] + vt0;
        ob[16] = acc11[r] + vt1;
    }
}

// ---------------------------------------------------------------------------
// Kernel 3: skinning. Per wave: Ts tile (16 verts x 16 comps) = W @ A[b,t],
// K=52 via 13 WMMAs. W fragments live in registers across 16 timesteps.
// A[b,t] is double-buffered in LDS via GLOBAL_LOAD_ASYNC_TO_LDS_B128
// (ASYNCcnt), so the copy of t+1 overlaps the WMMAs/apply of t.
// grid = (N/128, T/16, B) = (16, 8, 32), 256 threads (8 waves).
// ---------------------------------------------------------------------------
__device__ __forceinline__ void async_a_stage(const float* __restrict__ gsrc,
                                              const float* lds_dst, int tid) {
    // 832 floats = 208 x b128; threads 0..207 move 16B each.
    if (tid < (J_*16)/4) {
        unsigned lds_off = (unsigned)(size_t)(lds_dst + tid*4); // low 32b = LDS offset
        const float* g = gsrc + tid*4;
        asm volatile("global_load_async_to_lds_b128 %0, %1, off"
                     :: "v"(lds_off), "v"(g) : "memory");
    }
}

__device__ __forceinline__ void wait_async0() {
#if __has_builtin(__builtin_amdgcn_s_wait_asynccnt)
    __builtin_amdgcn_s_wait_asynccnt(0);
#else
    asm volatile("s_wait_asynccnt 0x0" ::: "memory");
#endif
}

__global__ void skin_kernel(const float* __restrict__ Amats,   // (B,T,52,16)
                            const float* __restrict__ lbsw,    // (B,2048,52)
                            const float* __restrict__ v_posed, // (B,T,6144)
                            const float* __restrict__ trans,   // (B,T,3)
                            float* __restrict__ out) {         // (B,T,2048,3)
    __shared__ __align__(16) float ldsA[2][J_ * 16];  // double-buffered A[b,t]
    __shared__ float ldsTs[8][16 * 17];               // per-wave Ts tile

    int tid   = threadIdx.x;
    int lane  = tid & 31;
    int w     = tid >> 5;
    int b     = blockIdx.z;
    int t0    = blockIdx.y * 16;
    int n0    = blockIdx.x * 128 + w * 16;
    int r15   = lane & 15;
    int khalf = (lane >> 4) * 2;

    // Preload W A-fragments for K=52 (13 steps), reused across 16 timesteps.
    v2f wf[13];
    const float* Wrow = lbsw + (size_t)(b*N_ + n0 + r15) * J_ + khalf;
    #pragma unroll
    for (int kk = 0; kk < 13; ++kk)
        wf[kk] = *(const v2f*)(Wrow + kk*4);

    // Prologue: start async copy of A[b,t0] into buffer 0.
    async_a_stage(Amats + (size_t)(b*T_ + t0) * (J_*16), ldsA[0], tid);

    for (int tt = 0; tt < 16; ++tt) {
        int t = t0 + tt;
        int cur = tt & 1;

        wait_async0();        // this wave's pending copy (buf cur) landed
        __syncthreads();      // all waves' pieces landed; prev readers of other buf done

        if (tt < 15)          // overlap copy of t+1 with compute of t
            async_a_stage(Amats + (size_t)(b*T_ + t + 1) * (J_*16),
                          ldsA[cur ^ 1], tid);

        const float* Abt = ldsA[cur];
        v8f acc = {0.f,0.f,0.f,0.f,0.f,0.f,0.f,0.f};
        #pragma unroll
        for (int kk = 0; kk < 13; ++kk) {
            int kr = kk*4 + khalf;
            v2f bb;
            bb.x = Abt[kr*16 + r15];
            bb.y = Abt[(kr+1)*16 + r15];
            acc = WMMA_F32(wf[kk], bb, acc);
        }

        // Scatter Ts tile: lane holds (vertex = r + 8*(lane>>4), comp = r15)
        #pragma unroll
        for (int r = 0; r < 8; ++r) {
            int vl = r + (lane >> 4) * 8;
            ldsTs[w][vl*17 + r15] = acc[r];
        }
        __syncthreads();

        // Apply 4x4 transform + trans; lanes 0..15 take one vertex each.
        if (lane < 16) {
            int n = n0 + lane;
            const float* Ts = &ldsTs[w][lane*17];
            const float* vp = v_posed + (size_t)(b*T_ + t) * M_ + n*3;
            const float* tr = trans + (size_t)(b*T_ + t) * 3;
            float x = vp[0], y = vp[1], z = vp[2];
            float* o = out + ((size_t)(b*T_ + t) * N_ + n) * 3;
            o[0] = Ts[0]*x + Ts[1]*y + Ts[2]*z  + Ts[3]  + tr[0];
            o[1] = Ts[4]*x + Ts[5]*y + Ts[6]*z  + Ts[7]  + tr[1];
            o[2] = Ts[8]*x + Ts[9]*y + Ts[10]*z + Ts[11] + tr[2];
        }
    }
}

// ---------------------------------------------------------------------------
extern "C" void kernel_launch(void* const* d_in, const int* in_sizes, int n_in,
                              void* d_out, int out_size, void* d_ws, size_t ws_size,
                              hipStream_t stream) {
    const float* pose_body  = (const float*)d_in[0];
    const float* trans      = (const float*)d_in[1];
    const float* betas      = (const float*)d_in[2];
    const float* A          = (const float*)d_in[3];
    const float* v_template = (const float*)d_in[4];
    const float* shapedirs  = (const float*)d_in[5];
    const float* posedirs   = (const float*)d_in[6];
    const float* lbsw       = (const float*)d_in[7];
    float* out = (float*)d_out;

    float* pf      = (float*)d_ws;                   // B*T*192 floats (~3 MB)
    float* v_posed = pf + (size_t)B_ * T_ * PFP;     // B*T*6144 floats (~101 MB)

    pf_kernel<<<(B_*T_*22 + 255)/256, 256, 0, stream>>>(pose_body, pf);
    vposed_kernel<<<dim3(24, 4, B_), 256, 0, stream>>>(pf, betas, posedirs,
                                                       shapedirs, v_template, v_posed);
    skin_kernel<<<dim3(16, 8, B_), 256, 0, stream>>>(A, lbsw, v_posed, trans, out);
}